// TemporalGNN_61495341744737
// MI455X (gfx1250) — compile-verified
//
#include <hip/hip_runtime.h>
#include <hip/hip_bf16.h>

// ---------------------------------------------------------------------------
// Problem constants (from reference)
// ---------------------------------------------------------------------------
#define BB 4
#define NN 10000
#define FF 8
#define TT 12
#define EE 160000
#define HH 512
#define FT (FF * TT)          // 96 floats per (b,n)

typedef __attribute__((ext_vector_type(2))) float v2f;
typedef __attribute__((ext_vector_type(8))) float v8f;

// FP32 WMMA 16x16x4:  D(16x16,f32) = A(16x4,f32) * B(4x16,f32) + C
// Layout (ISA 7.12.2):
//   A: lane m (=lane%16) holds row m; VGPR v, half h(=lane/16): K = v + 2h
//   B: lane m holds column m;        VGPR v, half h:            K = v + 2h
//   C/D: lane m holds column m;      VGPR v, half h: row M = v + 8h
__device__ __forceinline__ v8f wmma4(v2f a, v2f b, v8f c) {
  return __builtin_amdgcn_wmma_f32_16x16x4_f32(
      /*neg_a=*/false, a, /*neg_b=*/false, b,
      /*c_mod=*/(short)0, c, /*reuse_a=*/false, /*reuse_b=*/false);
}

__device__ __forceinline__ v8f v8splat(float x) {
  v8f r = {x, x, x, x, x, x, x, x};
  return r;
}

// Branch-free activations on the native transcendental unit (v_exp/v_rcp).
// one_minus_sigmoid(x) = 1/(1+e^x)
__device__ __forceinline__ float one_minus_sigmoid(float x) {
  return __builtin_amdgcn_rcpf(1.f + __expf(x));
}
// tanh(x) = 1 - 2/(e^(2x)+1); correct limits (+1/-1) without branches.
__device__ __forceinline__ float fast_tanh(float x) {
  return 1.f - 2.f * __builtin_amdgcn_rcpf(__expf(2.f * x) + 1.f);
}

// ---------------------------------------------------------------------------
// Prep kernels
// ---------------------------------------------------------------------------
__global__ void k_softmax_att(const float* __restrict__ att,
                              float* __restrict__ probs) {
  if (threadIdx.x == 0 && blockIdx.x == 0) {
    float mx = -1e30f;
    for (int t = 0; t < TT; ++t) mx = fmaxf(mx, att[t]);
    float e[TT], s = 0.f;
    for (int t = 0; t < TT; ++t) { e[t] = __expf(att[t] - mx); s += e[t]; }
    float inv = 1.f / s;
    for (int t = 0; t < TT; ++t) probs[t] = e[t] * inv;
  }
}

// Mz = Wz @ Wlz[0:512,:],  vz = bz @ Wlz[0:512,:] + blz   (same for h path)
__global__ void k_fold(const float* __restrict__ Wz, const float* __restrict__ bz,
                       const float* __restrict__ Wlz, const float* __restrict__ blz,
                       const float* __restrict__ Wh, const float* __restrict__ bh,
                       const float* __restrict__ Wlh, const float* __restrict__ blh,
                       float* __restrict__ Mz, float* __restrict__ vz,
                       float* __restrict__ Mh, float* __restrict__ vh) {
  int id = blockIdx.x * blockDim.x + threadIdx.x;
  if (id >= FF * HH) return;
  int f = id >> 9, c = id & (HH - 1);
  float sz = 0.f, sh = 0.f;
  for (int j = 0; j < HH; ++j) {
    sz += Wz[f * HH + j] * Wlz[j * HH + c];
    sh += Wh[f * HH + j] * Wlh[j * HH + c];
  }
  Mz[f * HH + c] = sz;
  Mh[f * HH + c] = sh;
  if (f == 0) {
    float tz = blz[c], th = blh[c];
    for (int j = 0; j < HH; ++j) {
      tz += bz[j] * Wlz[j * HH + c];
      th += bh[j] * Wlh[j * HH + c];
    }
    vz[c] = tz;
    vh[c] = th;
  }
}

__global__ void k_deg_init(float* __restrict__ deg) {
  int i = blockIdx.x * blockDim.x + threadIdx.x;
  if (i < NN) deg[i] = 1.0f;  // self-loop weight
}

__global__ void k_deg(const int* __restrict__ ei, const float* __restrict__ w,
                      float* __restrict__ deg) {
  int e = blockIdx.x * blockDim.x + threadIdx.x;
  if (e < EE) unsafeAtomicAdd(&deg[ei[EE + e]], w[e]);
}

__global__ void k_dinv(float* __restrict__ deg) {
  int i = blockIdx.x * blockDim.x + threadIdx.x;
  if (i < NN) {
    float d = deg[i];
    deg[i] = (d > 0.f) ? rsqrtf(d) : 0.f;
  }
}

__global__ void k_norm(const int* __restrict__ ei, const float* __restrict__ w,
                       const float* __restrict__ dinv, float* __restrict__ norm) {
  int e = blockIdx.x * blockDim.x + threadIdx.x;
  if (e < EE) norm[e] = dinv[ei[e]] * w[e] * dinv[ei[EE + e]];
}

// ax initialized with the self-loop contribution (fully overwrites workspace)
__global__ void k_selfloop(const float* __restrict__ x,
                           const float* __restrict__ dinv,
                           float* __restrict__ ax) {
  int id = blockIdx.x * blockDim.x + threadIdx.x;
  if (id < BB * NN * FT) {
    int n = (id / FT) % NN;
    float di = dinv[n];
    ax[id] = x[id] * di * di;
  }
}

// edge scatter: ax[b,dst,:] += norm * x[b,src,:]  (96 contiguous floats / node)
__global__ void k_scatter(const float* __restrict__ x, const int* __restrict__ ei,
                          const float* __restrict__ norm, float* __restrict__ ax) {
  int id = blockIdx.x * blockDim.x + threadIdx.x;
  if (id >= EE * FT) return;
  int e = id / FT, k = id - e * FT;
  int s = ei[e], d = ei[EE + e];
  float nm = norm[e];
  const int bstride = NN * FT;
  #pragma unroll
  for (int b = 0; b < BB; ++b)
    unsafeAtomicAdd(&ax[b * bstride + d * FT + k], x[b * bstride + s * FT + k] * nm);
}

// ---------------------------------------------------------------------------
// Fused temporal-gate + MLP kernel.  One block = 16 rows of (b*N+n).
// 128 threads = 4 waves; all WMMA in wave-uniform control flow, no EXEC
// divergence anywhere near the WMMA loops (branch-free activations).
// ---------------------------------------------------------------------------
#define AS 100   // a_lds row stride (96 + 4)  -> conflict free across 16 lanes
#define HS 516   // h_lds row stride (512 + 4) -> stride%64 == 4
#define Y1S 260  // 256 + 4
#define Y2S 132  // 128 + 4

__global__ __launch_bounds__(128) void k_fused(
    const float* __restrict__ ax,
    const float* __restrict__ Mz, const float* __restrict__ vz,
    const float* __restrict__ Mh, const float* __restrict__ vh,
    const float* __restrict__ probs,
    const float* __restrict__ W1, const float* __restrict__ b1,
    const float* __restrict__ W2, const float* __restrict__ b2,
    const float* __restrict__ W3, const float* __restrict__ b3,
    float* __restrict__ out) {
  __shared__ float a_lds[16 * AS];    // a[r][f*12+t]
  __shared__ float h_lds[16 * HS];
  __shared__ float y1_lds[16 * Y1S];
  __shared__ float y2_lds[16 * Y2S];
  __shared__ float p_lds[TT];

  const int tid = threadIdx.x;
  const int wave = tid >> 5;
  const int lane = tid & 31;
  const int m = lane & 15;   // A-row / B-col / C-col index
  const int hh = lane >> 4;  // lane half
  const long row0 = (long)blockIdx.x * 16;

  // Stage 16 rows x 96 contiguous floats of aggregated input
  for (int i = tid; i < 16 * FT; i += 128) {
    int r = i / FT, k = i - r * FT;
    a_lds[r * AS + k] = ax[(row0 + r) * FT + k];
  }
  if (tid < TT) p_lds[tid] = probs[tid];
  __syncthreads();

  // ---- Phase 1: h[16x512] = sum_t p_t * (1 - sigmoid(A_t Mz + vz)) * tanh(A_t Mh + vh)
  for (int i = 0; i < 8; ++i) {            // 32 column tiles / 4 waves
    int ct = wave * 8 + i;
    int c = ct * 16 + m;                   // global column 0..511
    float vzc = vz[c], vhc = vh[c];
    // B fragments are t-invariant: load once per column tile.
    v2f bz0 = {Mz[(2 * hh) * HH + c], Mz[(2 * hh + 1) * HH + c]};
    v2f bz1 = {Mz[(4 + 2 * hh) * HH + c], Mz[(5 + 2 * hh) * HH + c]};
    v2f bh0 = {Mh[(2 * hh) * HH + c], Mh[(2 * hh + 1) * HH + c]};
    v2f bh1 = {Mh[(4 + 2 * hh) * HH + c], Mh[(5 + 2 * hh) * HH + c]};
    float accf[8] = {0.f, 0.f, 0.f, 0.f, 0.f, 0.f, 0.f, 0.f};
    for (int t = 0; t < TT; ++t) {
      v2f a0 = {a_lds[m * AS + (2 * hh) * TT + t],
                a_lds[m * AS + (2 * hh + 1) * TT + t]};
      v2f a1 = {a_lds[m * AS + (4 + 2 * hh) * TT + t],
                a_lds[m * AS + (5 + 2 * hh) * TT + t]};
      v8f cz = v8splat(vzc);
      cz = wmma4(a0, bz0, cz);
      cz = wmma4(a1, bz1, cz);
      v8f ch = v8splat(vhc);
      ch = wmma4(a0, bh0, ch);
      ch = wmma4(a1, bh1, ch);
      float p = p_lds[t];
      #pragma unroll
      for (int j = 0; j < 8; ++j) {
        // p * (1 - sigmoid(cz)) * tanh(ch), fully branch-free
        accf[j] += p * one_minus_sigmoid(cz[j]) * fast_tanh(ch[j]);
      }
    }
    #pragma unroll
    for (int j = 0; j < 8; ++j) h_lds[(j + 8 * hh) * HS + c] = accf[j];
  }
  __syncthreads();

  // ---- Phase 2: y1 = relu(h @ W1 + b1)   (512 -> 256)
  for (int i = 0; i < 4; ++i) {            // 16 column tiles / 4 waves
    int ct = wave * 4 + i;
    int c = ct * 16 + m;                   // 0..255
    v8f acc = v8splat(b1[c]);
    for (int k0 = 0; k0 < HH; k0 += 4) {
      v2f a0 = {h_lds[m * HS + k0 + 2 * hh], h_lds[m * HS + k0 + 1 + 2 * hh]};
      v2f b0 = {W1[(k0 + 2 * hh) * 256 + c], W1[(k0 + 1 + 2 * hh) * 256 + c]};
      acc = wmma4(a0, b0, acc);
    }
    #pragma unroll
    for (int j = 0; j < 8; ++j)
      y1_lds[(j + 8 * hh) * Y1S + c] = fmaxf(acc[j], 0.f);
  }
  __syncthreads();

  // ---- Phase 3: y2 = relu(y1 @ W2 + b2)  (256 -> 128)
  for (int i = 0; i < 2; ++i) {            // 8 column tiles / 4 waves
    int ct = wave * 2 + i;
    int c = ct * 16 + m;                   // 0..127
    v8f acc = v8splat(b2[c]);
    for (int k0 = 0; k0 < 256; k0 += 4) {
      v2f a0 = {y1_lds[m * Y1S + k0 + 2 * hh], y1_lds[m * Y1S + k0 + 1 + 2 * hh]};
      v2f b0 = {W2[(k0 + 2 * hh) * 128 + c], W2[(k0 + 1 + 2 * hh) * 128 + c]};
      acc = wmma4(a0, b0, acc);
    }
    #pragma unroll
    for (int j = 0; j < 8; ++j)
      y2_lds[(j + 8 * hh) * Y2S + c] = fmaxf(acc[j], 0.f);
  }
  __syncthreads();

  // ---- Phase 4: out = y2 @ W3 + b3   (128 -> 1), 16 rows
  if (wave == 0 && lane < 16) {
    float s = b3[0];
    for (int k = 0; k < 128; ++k) s += y2_lds[lane * Y2S + k] * W3[k];
    out[row0 + lane] = s;
  }
}

// ---------------------------------------------------------------------------
// Host launch
// ---------------------------------------------------------------------------
extern "C" void kernel_launch(void* const* d_in, const int* in_sizes, int n_in,
                              void* d_out, int out_size, void* d_ws, size_t ws_size,
                              hipStream_t stream) {
  const float* x   = (const float*)d_in[0];
  const int*   ei  = (const int*)d_in[1];
  const float* w   = (const float*)d_in[2];
  const float* att = (const float*)d_in[3];
  const float* Wz  = (const float*)d_in[4];
  const float* bz  = (const float*)d_in[5];
  // d_in[6], d_in[7] = Wr, br  : dead (H0 == 0)
  const float* Wh  = (const float*)d_in[8];
  const float* bh  = (const float*)d_in[9];
  const float* Wlz = (const float*)d_in[10];
  const float* blz = (const float*)d_in[11];
  // d_in[12], d_in[13] = Wlr, blr : dead
  const float* Wlh = (const float*)d_in[14];
  const float* blh = (const float*)d_in[15];
  const float* W1  = (const float*)d_in[16];
  const float* b1  = (const float*)d_in[17];
  const float* W2  = (const float*)d_in[18];
  const float* b2  = (const float*)d_in[19];
  const float* W3  = (const float*)d_in[20];
  const float* b3  = (const float*)d_in[21];
  float* out = (float*)d_out;

  // Workspace layout (floats), ~16.1 MB total
  float* ws    = (float*)d_ws;
  float* probs = ws + 0;        // 12
  float* vz    = ws + 16;       // 512
  float* vh    = ws + 528;      // 512
  float* Mz    = ws + 1040;     // 4096
  float* Mh    = ws + 5136;     // 4096
  float* dinv  = ws + 9232;     // 10000 (deg, then in-place 1/sqrt(deg))
  float* norm  = ws + 19232;    // 160000
  float* ax    = ws + 179232;   // 3,840,000

  k_softmax_att<<<1, 1, 0, stream>>>(att, probs);
  k_fold<<<(FF * HH + 255) / 256, 256, 0, stream>>>(Wz, bz, Wlz, blz,
                                                    Wh, bh, Wlh, blh,
                                                    Mz, vz, Mh, vh);
  k_deg_init<<<(NN + 255) / 256, 256, 0, stream>>>(dinv);
  k_deg<<<(EE + 255) / 256, 256, 0, stream>>>(ei, w, dinv);
  k_dinv<<<(NN + 255) / 256, 256, 0, stream>>>(dinv);
  k_norm<<<(EE + 255) / 256, 256, 0, stream>>>(ei, w, dinv, norm);
  k_selfloop<<<(BB * NN * FT + 255) / 256, 256, 0, stream>>>(x, dinv, ax);
  k_scatter<<<(EE * FT + 255) / 256, 256, 0, stream>>>(x, ei, norm, ax);
  k_fused<<<(BB * NN) / 16, 128, 0, stream>>>(ax, Mz, vz, Mh, vh, probs,
                                              W1, b1, W2, b2, W3, b3, out);
}